// Four_Pos_Fusion_Embedding_2834678415818
// MI455X (gfx1250) — compile-verified
//
#include <hip/hip_runtime.h>

// Problem constants (match the reference)
#define BDIM   2
#define SDIM   256
#define HDIM   256
#define MAXL   512
#define PEROWS 1025           // 2*MAXL + 1 rows in the PE table

typedef __attribute__((ext_vector_type(2))) float v2f;
typedef __attribute__((ext_vector_type(4))) float v4f;
typedef __attribute__((ext_vector_type(8))) float v8f;

// ---------------------------------------------------------------------------
// Stage 1: PEW[k] = pe @ W[k*H:(k+1)*H]   (k = 0..3)
//   pe:  [1025, 256] f32 row-major
//   W:   [1024, 256] f32 row-major (4 stacked 256x256 blocks, K x N layout)
//   pew: [4, 1025, 256] f32 in workspace
// One wave (32 threads) per 16x16 output tile, K=256 in steps of 4 via
// V_WMMA_F32_16X16X4_F32. Total 65*16*4 = 4160 waves, 64 WMMA each.
// ---------------------------------------------------------------------------
__global__ __launch_bounds__(32) void pew_gemm_kernel(
    const float* __restrict__ pe,
    const float* __restrict__ W,
    float* __restrict__ pew)
{
  int tile = blockIdx.x;                 // 0 .. 65*16*4 - 1
  int k    = tile / (65 * 16);           // which 256-row block of W
  int rem  = tile - k * (65 * 16);
  int tm   = rem >> 4;                   // M tile (0..64)
  int tn   = rem & 15;                   // N tile (0..15)

  int lane = threadIdx.x;                // wave32
  int hi   = lane >> 4;                  // 0/1 : which K-halfpair this lane holds
  int l16  = lane & 15;

  // A (16x4, MxK): lane holds row M=l16; VGPR0/1 = K = {2*hi, 2*hi+1} (+ks)
  int arow = tm * 16 + l16;
  if (arow > PEROWS - 1) arow = PEROWS - 1;          // clamp tail rows
  const float* aptr = pe + (size_t)arow * HDIM + 2 * hi;

  // B (4x16, KxN): row K striped across lanes in one VGPR:
  //   VGPR0 = B[ks + 2*hi][tn*16 + l16], VGPR1 = B[ks + 2*hi + 1][...]
  const float* bptr = W + (size_t)k * HDIM * HDIM
                        + (size_t)(2 * hi) * HDIM + tn * 16 + l16;

  v8f acc = {};
#pragma unroll 8
  for (int ks = 0; ks < HDIM; ks += 4) {
    v2f a = *(const v2f*)(aptr + ks);    // contiguous pair, 8B aligned
    v2f b;
    b.x = bptr[(size_t)ks * HDIM];
    b.y = bptr[(size_t)ks * HDIM + HDIM];
    acc = __builtin_amdgcn_wmma_f32_16x16x4_f32(
        /*neg_a=*/false, a, /*neg_b=*/false, b,
        /*c_mod=*/(short)0, acc, /*reuse_a=*/false, /*reuse_b=*/false);
  }

  // D (16x16 f32): lane l -> N = l16; VGPR g -> M = g + 8*hi
  float* dbase = pew + (size_t)k * PEROWS * HDIM + tn * 16 + l16;
#pragma unroll
  for (int g = 0; g < 8; ++g) {
    int m = tm * 16 + g + 8 * hi;
    if (m < PEROWS) dbase[(size_t)m * HDIM] = acc[g];
  }
}

// ---------------------------------------------------------------------------
// Stage 2: out[b,i,j,:] = lrelu( PEW0[ss] + PEW1[se] + PEW2[es] + PEW3[ee] + b )
// 4 gathered b128 loads (L2-resident tables) + bias, nontemporal b128 store.
// Block = 256 threads: 4 j-values x 64 threads (64 x float4 = 256 floats/row).
// Grid = B * S * (S/4) = 32768 blocks.
// ---------------------------------------------------------------------------
__global__ __launch_bounds__(256) void gather_bias_lrelu_kernel(
    const int* __restrict__ pos_s,
    const int* __restrict__ pos_e,
    const float* __restrict__ pew,
    const float* __restrict__ bias,
    float* __restrict__ out)
{
  int blk = blockIdx.x;
  int jt  = blk & 63;                    // j tile (4 j's each)
  int bi  = blk >> 6;                    // b*S + i  (uniform -> scalar loads)
  int b   = bi >> 8;

  int tid = threadIdx.x;
  int jl  = tid >> 6;                    // 0..3
  int j   = (jt << 2) + jl;
  int h   = (tid & 63) << 2;             // float4 offset within H

  int psi = pos_s[bi];
  int pei = pos_e[bi];
  int bj  = (b << 8) + j;
  int psj = pos_s[bj];
  int pej = pos_e[bj];

  // row indices into the 4 stacked PEW tables (each diff+MAXL in [1,1023])
  size_t iss = (size_t)(psi - psj + MAXL);
  size_t ise = (size_t)(PEROWS     + (psi - pej + MAXL));
  size_t ies = (size_t)(2 * PEROWS + (pei - psj + MAXL));
  size_t iee = (size_t)(3 * PEROWS + (pei - pej + MAXL));

  v4f r0 = *(const v4f*)(pew + iss * HDIM + h);
  v4f r1 = *(const v4f*)(pew + ise * HDIM + h);
  v4f r2 = *(const v4f*)(pew + ies * HDIM + h);
  v4f r3 = *(const v4f*)(pew + iee * HDIM + h);
  v4f bb = *(const v4f*)(bias + h);

  v4f v   = r0 + r1 + r2 + r3 + bb;
  v4f neg = v * 0.01f;
  v4f res;
  res.x = v.x > 0.0f ? v.x : neg.x;
  res.y = v.y > 0.0f ? v.y : neg.y;
  res.z = v.z > 0.0f ? v.z : neg.z;
  res.w = v.w > 0.0f ? v.w : neg.w;

  size_t off = ((size_t)bi * SDIM + j) * HDIM + h;
  __builtin_nontemporal_store(res, (v4f*)(out + off));   // keep PEW hot in L2
}

// ---------------------------------------------------------------------------
extern "C" void kernel_launch(void* const* d_in, const int* in_sizes, int n_in,
                              void* d_out, int out_size, void* d_ws, size_t ws_size,
                              hipStream_t stream) {
  (void)in_sizes; (void)n_in; (void)out_size; (void)ws_size;

  const int*   pos_s = (const int*)d_in[0];     // [B,S] int32
  const int*   pos_e = (const int*)d_in[1];     // [B,S] int32
  const float* pe    = (const float*)d_in[2];   // [1025,256] f32
  const float* W     = (const float*)d_in[3];   // [1024,256] f32
  const float* bias  = (const float*)d_in[4];   // [256] f32
  float*       out   = (float*)d_out;           // [B,S,S,H] f32
  float*       pew   = (float*)d_ws;            // 4*1025*256*4 = 4,198,400 B

  // Stage 1: 65 M-tiles x 16 N-tiles x 4 blocks, one wave each.
  pew_gemm_kernel<<<65 * 16 * 4, 32, 0, stream>>>(pe, W, pew);

  // Stage 2: bandwidth pass (stream-ordered after stage 1).
  gather_bias_lrelu_kernel<<<BDIM * SDIM * (SDIM / 4), 256, 0, stream>>>(
      pos_s, pos_e, pew, bias, out);
}